// SoftPathMDDLoss_9783935500402
// MI455X (gfx1250) — compile-verified
//
#include <hip/hip_runtime.h>
#include <hip/hip_bf16.h>

// ---------------------------------------------------------------------------
// SoftPathMDDLoss, reformulated as a chunk-parallel affine scan.
//   Q_t = 1 + Q_{t-1} * g_t,  g_t = (1+r_t)^(-BETA)
//   S   = sum_t Q_t,  loss_b = LAMBDA * relu(1 - S^(-1/BETA) - 0.12)
// All running quantities kept as (float mantissa, int exponent) pairs,
// renormalized every 16 elements -> 2 transcendentals + ~7 VALU per element.
// Single pass over the 256MB input (HBM roofline ~11us @ 23.3 TB/s).
// ---------------------------------------------------------------------------

#define T_STEPS 65536
#define B_COLS  1024
#define NCHUNK  128                  // chunks along T
#define TC      (T_STEPS / NCHUNK)   // 512 rows per chunk
#define TT      32                   // tile rows staged in LDS per buffer
#define NTILES  (TC / TT)            // 16 tiles per chunk
#define CBLK    256                  // columns per workgroup (== blockDim.x)
#define NCB     (B_COLS / CBLK)      // 4 column blocks
#define XFERS   8                    // async b128 transfers per thread per tile

#if defined(__has_builtin)
# if __has_builtin(__builtin_amdgcn_global_load_async_to_lds_b128)
#  define USE_ASYNC_COPY 1
# endif
# if __has_builtin(__builtin_amdgcn_s_wait_asynccnt)
#  define HAVE_WAIT_ASYNC_BUILTIN 1
# endif
#endif

typedef int v4i __attribute__((vector_size(16)));   // matches builtin param type

template <int N>
__device__ __forceinline__ void wait_async_le() {
#ifdef USE_ASYNC_COPY
# ifdef HAVE_WAIT_ASYNC_BUILTIN
  __builtin_amdgcn_s_wait_asynccnt((unsigned short)N);
# else
  asm volatile("s_wait_asynccnt %0" ::"i"(N) : "memory");
# endif
#endif
}

__device__ __forceinline__ void renorm(float& m, int& e) {
  int k;
  m = frexpf(m, &k);   // v_frexp_mant_f32 / v_frexp_exp_i32 (cheap VALU)
  e += k;
}

// (sm * 2^se) += (tm * 2^te), keeping sm normalized.
__device__ __forceinline__ void add_scaled(float& sm, int& se, float tm, int te) {
  if (tm == 0.0f) return;
  if (sm == 0.0f) {
    sm = tm; se = te;
  } else {
    int d = te - se;
    if (d > 0) { sm = ldexpf(sm, -d) + tm; se = te; }
    else       { sm += ldexpf(tm, d); }           // d<=0; huge |d| underflows to 0 (ok)
  }
  renorm(sm, se);
}

// Stage one TT x CBLK tile (32KB) into LDS. Wave-contiguous 512B per issue.
__device__ __forceinline__ void stage_tile(const float* __restrict__ in,
                                           size_t tile_elem0, float* lbuf, int tid) {
#pragma unroll
  for (int j = 0; j < XFERS; ++j) {
    const int k       = j * CBLK + tid;       // 0..2047
    const int row     = k >> 6;               // 64 x 16B spans one 1KB tile row
    const int colbyte = (k & 63) * 16;
    const char* g = (const char*)in +
                    (tile_elem0 + (size_t)row * B_COLS) * sizeof(float) + colbyte;
    char* l = (char*)lbuf + (size_t)k * 16;
#ifdef USE_ASYNC_COPY
    __builtin_amdgcn_global_load_async_to_lds_b128(
        (__attribute__((address_space(1))) v4i*)g,
        (__attribute__((address_space(3))) v4i*)l,
        0, 0);
#else
    *(float4*)l = *(const float4*)g;          // synchronous fallback
#endif
  }
}

// ---------------------------------------------------------------------------
// Pass 1: each workgroup owns (chunk, column-block); computes per column:
//   sumL2 = sum log2(1+r)                    (=> chunk product A = 2^(-40*sumL2))
//   B  = local Q at chunk end with Q_in = 0          (scaled)
//   S0 = local sum of Q_t with Q_in = 0              (scaled)
//   G  = sum_t prod_{u<=t} g_u  (coefficient of Q_in in S)   (scaled)
// ---------------------------------------------------------------------------
__global__ __launch_bounds__(CBLK) void mdd_pass1(const float* __restrict__ in,
                                                  float* __restrict__ ws) {
  __shared__ float tile[2][TT * CBLK];        // 2 x 32KB double buffer

  const int tid   = threadIdx.x;
  const int cb    = blockIdx.x & (NCB - 1);
  const int chunk = blockIdx.x >> 2;          // log2(NCB) == 2
  const int t0    = chunk * TC;
  const size_t base = (size_t)t0 * B_COLS + (size_t)cb * CBLK;

  // scaled scan state
  float qm = 0.0f, sm = 0.0f, rm = 1.0f, gm = 0.0f, suml2 = 0.0f;
  int   qe = 0,    se = 0,    re = 0,    ge = 0;

  stage_tile(in, base, &tile[0][0], tid);

  for (int ti = 0; ti < NTILES; ++ti) {
    if (ti + 1 < NTILES)
      stage_tile(in, base + (size_t)(ti + 1) * TT * B_COLS, &tile[(ti + 1) & 1][0], tid);
    if (ti + 1 < NTILES) wait_async_le<XFERS>();   // oldest tile (8 issues/wave) done
    else                 wait_async_le<0>();
    __syncthreads();

    const float* buf = &tile[ti & 1][0];
#pragma unroll
    for (int sb = 0; sb < TT; sb += 16) {
      // exponents frozen for 16 steps -> pure FMA inner loop
      const float p1  = ldexpf(1.0f, -qe);       // "+1" in Q's scale
      const float wsq = ldexpf(1.0f, qe - se);   // Q -> S weight
      const float wg  = ldexpf(1.0f, re - ge);   // R -> G weight
#pragma unroll
      for (int r2 = 0; r2 < 16; ++r2) {
        const float r = buf[(sb + r2) * CBLK + tid];
        const float l = __log2f(1.0f + r);       // v_log_f32
        suml2 += l;
        const float g = exp2f(-40.0f * l);       // v_exp_f32 : (1+r)^-BETA
        qm = __builtin_fmaf(qm, g, p1);          // Q = 1 + Q*g
        sm = __builtin_fmaf(qm, wsq, sm);        // S += Q
        rm = rm * g;                             // R = prod g
        gm = __builtin_fmaf(rm, wg, gm);         // G += R
      }
      renorm(qm, qe); renorm(sm, se); renorm(rm, re); renorm(gm, ge);
    }
    __syncthreads();                             // buffer safe for next async fill
  }

  const size_t N   = (size_t)NCHUNK * B_COLS;
  const size_t idx = (size_t)chunk * B_COLS + (size_t)cb * CBLK + tid;
  ws[0 * N + idx] = suml2;
  ws[1 * N + idx] = qm;  ws[2 * N + idx] = (float)qe;
  ws[3 * N + idx] = sm;  ws[4 * N + idx] = (float)se;
  ws[5 * N + idx] = gm;  ws[6 * N + idx] = (float)ge;
}

// ---------------------------------------------------------------------------
// Pass 2: per column, fold the NCHUNK affine summaries in order, then loss.
//   S_tot += S0_c + Q_in * G_c ;  Q_in = B_c + Q_in * A_c
// 3.5MB of summaries, fully L2-resident.
// ---------------------------------------------------------------------------
__global__ __launch_bounds__(CBLK) void mdd_pass2(const float* __restrict__ ws,
                                                  float* __restrict__ out) {
  const int b = blockIdx.x * CBLK + threadIdx.x;   // 0..1023
  const size_t N = (size_t)NCHUNK * B_COLS;

  float Qm = 0.0f, Sm = 0.0f;
  int   Qe = 0,    Se = 0;

  for (int c = 0; c < NCHUNK; ++c) {
    const size_t idx = (size_t)c * B_COLS + b;
    const float suml2 = ws[0 * N + idx];
    const float bqm   = ws[1 * N + idx];
    const int   bqe   = (int)ws[2 * N + idx];
    const float scm   = ws[3 * N + idx];
    const int   sce   = (int)ws[4 * N + idx];
    const float ggm   = ws[5 * N + idx];
    const int   gge   = (int)ws[6 * N + idx];

    // S_tot += S0_c
    add_scaled(Sm, Se, scm, sce);
    // S_tot += Q_in * G_c
    {
      float tm = Qm * ggm; int te = Qe + gge;
      if (tm != 0.0f) { renorm(tm, te); add_scaled(Sm, Se, tm, te); }
    }
    // Q_in = B_c + Q_in * A_c,  A_c = 2^(-40*suml2)
    {
      const float x  = -40.0f * suml2;
      const float xf = floorf(x);
      const float am = exp2f(x - xf);
      const int   ae = (int)xf;
      float pm = Qm * am; int pe = Qe + ae;
      float nm = bqm;     int ne = bqe;
      if (pm != 0.0f) { renorm(pm, pe); add_scaled(nm, ne, pm, pe); }
      Qm = nm; Qe = ne;
    }
  }

  // soft_mdd_log = ln(S)/BETA ; mdd = 1 - S^(-1/BETA)
  const float log2S = __log2f(Sm) + (float)Se;     // S >= T so log2S >= 16
  const float mdd   = 1.0f - exp2f(-log2S * (1.0f / 40.0f));
  const float hinge = fmaxf(mdd - 0.12f, 0.0f);    // threshold = 0.1 + 0.02
  out[b] = 5.0f * hinge;                           // MDD_LAMBDA
}

// ---------------------------------------------------------------------------
extern "C" void kernel_launch(void* const* d_in, const int* in_sizes, int n_in,
                              void* d_out, int out_size, void* d_ws, size_t ws_size,
                              hipStream_t stream) {
  (void)in_sizes; (void)n_in; (void)out_size; (void)ws_size;
  const float* in = (const float*)d_in[0];
  float* ws  = (float*)d_ws;    // needs 7 * NCHUNK * B_COLS * 4 = 3.5 MB
  float* out = (float*)d_out;

  mdd_pass1<<<dim3(NCHUNK * NCB), dim3(CBLK), 0, stream>>>(in, ws);
  mdd_pass2<<<dim3(B_COLS / CBLK), dim3(CBLK), 0, stream>>>(ws, out);
}